// EmbeddingToExpression_45157286150933
// MI455X (gfx1250) — compile-verified
//
#include <hip/hip_runtime.h>

typedef __attribute__((ext_vector_type(2))) float v2f;
typedef __attribute__((ext_vector_type(8))) float v8f;

#define NR   1024   // regions
#define NC   8192   // cells
#define DIN  16
#define NE   5      // hidden E
#define CHUNKS 8    // blockIdx.y chunks over cells
#define CELLS_PER_CHUNK (NC / CHUNKS)               // 1024
#define TILES_PER_WAVE  (CELLS_PER_CHUNK / 16 / 8)  // 8

// Branch-free exact-GELU: x * 0.5 * (1 + erf(x/sqrt(2))) with
// Abramowitz-Stegun 7.1.26 erf (|abs err| <= 1.5e-7), v_rcp + v_exp, no
// EXEC divergence. Saturates correctly for large |x| via exp underflow.
__device__ __forceinline__ float gelu_exact(float x) {
    const float z = 0.70710678118654752440f * x;
    const float s = fabsf(z);
    const float t = __builtin_amdgcn_rcpf(fmaf(0.3275911f, s, 1.0f));
    float poly = fmaf(1.061405429f, t, -1.453152027f);
    poly = fmaf(poly, t, 1.421413741f);
    poly = fmaf(poly, t, -0.284496736f);
    poly = fmaf(poly, t, 0.254829592f);
    poly *= t;
    const float e = __expf(-s * s);
    const float erf_abs = fmaf(-poly, e, 1.0f);
    const float erfz = copysignf(erf_abs, z);
    return 0.5f * x * (1.0f + erfz);
}

__global__ __launch_bounds__(256)
void EmbeddingToExpression_45157286150933_kernel(
        const float* __restrict__ emb,   // [R, C, DIN]
        const int*   __restrict__ oi,    // [R]
        const float* __restrict__ W1,    // [NREG, DIN, E]
        const float* __restrict__ b1,    // [NREG, E]
        const float* __restrict__ Wf,    // [NREG, E, 1]
        float*       __restrict__ out)   // [C, R]
{
    const int r     = blockIdx.x;        // region
    const int chunk = blockIdx.y;        // cell chunk
    const int tid   = threadIdx.x;
    const int wave  = tid >> 5;
    const int lane  = tid & 31;
    const int m     = lane & 15;         // A: row (=E index); B/D: column (=cell)
    const int hh    = lane >> 4;         // which 16-lane half

    const int g = oi[r];                 // gathered weight row

    // ---- A fragments: W1^T, 16x4 f32 per WMMA. Row M = E index (rows >=5
    //      zero); lane layout: VGPR0 = {K=kb*4+0 | +2}, VGPR1 = {+1 | +3}.
    const float* w1g = W1 + (size_t)g * (DIN * NE);
    v2f Af[4];
#pragma unroll
    for (int kb = 0; kb < 4; ++kb) {
        const int k0 = kb * 4 + 2 * hh;
        const float x0 = (m < NE) ? w1g[(k0 + 0) * NE + m] : 0.0f;
        const float x1 = (m < NE) ? w1g[(k0 + 1) * NE + m] : 0.0f;
        v2f a = { x0, x1 };
        Af[kb] = a;
    }

    // Per-(j,half) epilogue constants: e = j + 8*hh; zero for e >= 5.
    const float* b1g = b1 + (size_t)g * NE;
    const float* wfg = Wf + (size_t)g * NE;
    float bj[NE], wfj[NE];
#pragma unroll
    for (int j = 0; j < NE; ++j) {
        bj[j]  = (hh == 0) ? b1g[j] : 0.0f;
        wfj[j] = (hh == 0) ? wfg[j] : 0.0f;
    }

    // Base of this block's cell chunk inside region r (contiguous f32 stream)
    const float* embr = emb + ((size_t)r * NC + (size_t)chunk * CELLS_PER_CHUNK) * DIN;

    for (int i = 0; i < TILES_PER_WAVE; ++i) {
        const int    tl    = i * 8 + wave;             // local 16-cell tile
        const float* tileA = embr + (size_t)tl * 16 * DIN;

        // B fragments: X^T (4x16 per WMMA), N = cell = lane&15. Four float2
        // loads per lane; lane pair (l, l+16) fully consumes each 64B row.
        v2f B0 = *(const v2f*)(tileA + m * DIN + 0 * 4 + 2 * hh);
        v2f B1 = *(const v2f*)(tileA + m * DIN + 1 * 4 + 2 * hh);
        v2f B2 = *(const v2f*)(tileA + m * DIN + 2 * 4 + 2 * hh);
        v2f B3 = *(const v2f*)(tileA + m * DIN + 3 * 4 + 2 * hh);

        v8f acc = {};
        acc = __builtin_amdgcn_wmma_f32_16x16x4_f32(false, Af[0], false, B0,
                                                    (short)0, acc, false, false);
        acc = __builtin_amdgcn_wmma_f32_16x16x4_f32(false, Af[1], false, B1,
                                                    (short)0, acc, false, false);
        acc = __builtin_amdgcn_wmma_f32_16x16x4_f32(false, Af[2], false, B2,
                                                    (short)0, acc, false, false);
        acc = __builtin_amdgcn_wmma_f32_16x16x4_f32(false, Af[3], false, B3,
                                                    (short)0, acc, false, false);

        // D layout: acc[j] = h[e = j + 8*hh][cell = lane&15].
        // Stage 2 is a per-lane dot over j — no cross-lane reduction needed.
        float p = 0.0f;
#pragma unroll
        for (int j = 0; j < NE; ++j) {
            const float hv = acc[j] + bj[j];           // exact 0 in upper half
            p = fmaf(gelu_exact(hv), wfj[j], p);       // wfj==0 kills e>=5
        }

        const int cellbase = chunk * CELLS_PER_CHUNK + tl * 16;
        if (lane < 16) {
            out[(size_t)(cellbase + lane) * NR + r] = p;   // [C, R] store
        }
    }
}

extern "C" void kernel_launch(void* const* d_in, const int* in_sizes, int n_in,
                              void* d_out, int out_size, void* d_ws, size_t ws_size,
                              hipStream_t stream) {
    const float* emb = (const float*)d_in[0];   // region_cell_embedding [R,C,DIN]
    const int*   oi  = (const int*)  d_in[1];   // regions_oi [R]
    const float* W1  = (const float*)d_in[2];   // [NREG, DIN, E]
    const float* b1  = (const float*)d_in[3];   // [NREG, E]
    const float* Wf  = (const float*)d_in[4];   // [NREG, E, 1]
    float* out = (float*)d_out;                 // [C, R]

    dim3 grid(NR, CHUNKS);
    EmbeddingToExpression_45157286150933_kernel<<<grid, 256, 0, stream>>>(
        emb, oi, W1, b1, Wf, out);
}